// HexPlaneField_69071664054865
// MI455X (gfx1250) — compile-verified
//
#include <hip/hip_runtime.h>
#include <stdint.h>

// ---------------------------------------------------------------------------
// HexPlaneField forward for MI455X (gfx1250, wave32).
// - Ones-planes (any combo touching dim 3) computed analytically.
// - Real grids transposed CHW->HWC into d_ws so lane==channel gives one
//   coalesced 128B request per bilinear corner (wave32 == 32 channels).
// - Point stream staged into LDS with the Tensor Data Mover (double-buffered,
//   s_wait_tensorcnt pipelining); cooperative-load fallback if TDM builtins
//   are unavailable.
// - Output (512MB) written with non-temporal stores to keep the ~128MB grid
//   working set resident in the 192MB L2.
// ---------------------------------------------------------------------------

constexpr int BATCH  = 128;  // points staged per TDM transfer
constexpr int KBATCH = 4;    // batches per block
constexpr int WAVES  = 8;    // 256 threads / wave32
constexpr int PPW2   = BATCH / WAVES;  // points per wave per batch

struct GridPtrs { const float* g[12]; };

// ------------------------- TDM availability probes -------------------------
#if defined(__HIP_DEVICE_COMPILE__)
#if __has_builtin(__builtin_amdgcn_tensor_load_to_lds) && \
    __has_builtin(__builtin_amdgcn_s_wait_tensorcnt)
#define USE_TDM 1
#if __has_include(<hip/amd_detail/amd_gfx1250_TDM.h>)
#define TDM_6ARG 1  // amdgpu-toolchain (clang-23 / therock-10.0 headers)
#endif
#endif
#endif

#ifdef USE_TDM
typedef unsigned int tdm_v4u __attribute__((ext_vector_type(4)));
typedef int          tdm_v8i __attribute__((ext_vector_type(8)));
typedef int          tdm_v4i __attribute__((ext_vector_type(4)));

// 1-D DMA: nelem 4-byte elements from gsrc -> LDS byte address lds_byte_addr.
// D# per CDNA5 ISA ch.8: group0 {count=1, lds_addr, global_addr, type=2},
// group1 {data_size=4B, tensor_dim0=tile_dim0=nelem, tile_dim1=0 => 1-D}.
__device__ __forceinline__ void tdm_load_1d(const void* gsrc,
                                            unsigned lds_byte_addr,
                                            unsigned nelem) {
  const unsigned long long ga = (unsigned long long)(uintptr_t)gsrc;
  tdm_v4u g0;
  g0.x = 1u;                                             // count=1 (valid D#)
  g0.y = lds_byte_addr;                                  // bits 63:32
  g0.z = (unsigned)(ga & 0xffffffffu);                   // global_addr[31:0]
  g0.w = (unsigned)((ga >> 32) & 0x01ffffffu)            // global_addr[56:32]
       | (2u << 30);                                     // type=2 ("image")
  tdm_v8i g1;
  g1[0] = (int)(2u << 16);                               // data_size=4B
  g1[1] = (int)((nelem & 0xffffu) << 16);                // tensor_dim0[15:0]
  g1[2] = (int)((nelem >> 16) & 0xffffu);                // tensor_dim0[31:16]
  g1[3] = (int)((nelem & 0xffffu) << 16);                // tile_dim0
  g1[4] = 0;                                             // tile_dim1/2 = 0
  g1[5] = (int)nelem;                                    // dim0_stride[31:0]
  g1[6] = 0;
  g1[7] = 0;
  tdm_v4i z4 = (tdm_v4i)0;
#ifdef TDM_6ARG
  tdm_v8i z8 = (tdm_v8i)0;
  __builtin_amdgcn_tensor_load_to_lds(g0, g1, z4, z4, z8, 0);
#else
  __builtin_amdgcn_tensor_load_to_lds(g0, g1, z4, z4, 0);
#endif
}
#endif  // USE_TDM

// -------------------------- transpose pre-pass -----------------------------
// (32, H, W) -> (H, W, 32). Block = 256 threads, tile = 32c x 64w, one h row.
__global__ __launch_bounds__(256) void transpose_chw_to_hwc(
    const float* __restrict__ src, float* __restrict__ dst, int H, int W) {
  __shared__ float tile[32][65];
  const int h  = blockIdx.y;
  const int w0 = blockIdx.x * 64;
  const int tid = threadIdx.x;
#pragma unroll
  for (int r = 0; r < 8; ++r) {
    const int c  = r * 4 + (tid >> 6);
    const int wi = tid & 63;
    tile[c][wi] = src[(size_t)c * H * W + (size_t)h * W + (w0 + wi)];
  }
  __syncthreads();
#pragma unroll
  for (int r = 0; r < 8; ++r) {
    const int wi = r * 8 + (tid >> 5);
    const int c  = tid & 31;
    dst[((size_t)h * W + (w0 + wi)) * 32 + c] = tile[c][wi];
  }
}

// ------------------------------ staging ------------------------------------
__device__ __forceinline__ void stage_batch(float (&sbuf)[2][BATCH * 4],
                                            int buf, int bi,
                                            const float* __restrict__ pts,
                                            const float* __restrict__ joints,
                                            int n) {
#ifdef USE_TDM
  if (threadIdx.x < 32) {  // wave 0 issues the DMA (TDM ignores EXEC)
    const unsigned ldsBase = (unsigned)(uintptr_t)&sbuf[buf][0];
    const int e0 = bi * BATCH * 3;
    const unsigned np = (unsigned)min(BATCH * 3, 3 * n - e0);
    tdm_load_1d(pts + e0, ldsBase, np);
    const int j0 = bi * BATCH;
    const unsigned nj = (unsigned)min(BATCH, n - j0);
    tdm_load_1d(joints + j0, ldsBase + BATCH * 3 * 4, nj);
  }
#else
  const int t = threadIdx.x;
#pragma unroll
  for (int r = 0; r < 2; ++r) {
    const int idx = t + r * 256;  // 0..511
    if (idx < BATCH * 3) {
      const int g = bi * BATCH * 3 + idx;
      sbuf[buf][idx] = (g < 3 * n) ? pts[g] : 0.0f;
    } else {
      const int j = idx - BATCH * 3;
      const int g = bi * BATCH + j;
      sbuf[buf][BATCH * 3 + j] = (g < n) ? joints[g] : 0.0f;
    }
  }
#endif
}

// ----------------------------- main kernel ---------------------------------
template <bool TRANSPOSED>
__device__ __forceinline__ float bilin_eval(const float* __restrict__ g,
                                            int lane, int R,
                                            int x0, int x1, float wx,
                                            int y0, int y1, float wy) {
  float v00, v01, v10, v11;
  if (TRANSPOSED) {
    v00 = g[(y0 * R + x0) * 32 + lane];
    v01 = g[(y0 * R + x1) * 32 + lane];
    v10 = g[(y1 * R + x0) * 32 + lane];
    v11 = g[(y1 * R + x1) * 32 + lane];
  } else {
    const size_t cb = (size_t)lane * R * R;
    v00 = g[cb + y0 * R + x0];
    v01 = g[cb + y0 * R + x1];
    v10 = g[cb + y1 * R + x0];
    v11 = g[cb + y1 * R + x1];
  }
  const float ix0 = v00 * (1.0f - wx) + v01 * wx;
  const float ix1 = v10 * (1.0f - wx) + v11 * wx;
  return ix0 * (1.0f - wy) + ix1 * wy;
}

template <bool TRANSPOSED>
__global__ __launch_bounds__(256) void hexplane_kernel(
    const float* __restrict__ pts, const float* __restrict__ joints,
    float* __restrict__ out, int n, int nBatches, GridPtrs gp) {
  __shared__ float sbuf[2][BATCH * 4];  // [pts: 3*BATCH][joints: BATCH] x2
  const int tid  = threadIdx.x;
  const int lane = tid & 31;
  const int wave = tid >> 5;
  const float SCALE = (float)(2.0 / (-2.0 * 1.3));

  const int b0 = blockIdx.x * KBATCH;
  const int kmax = min(KBATCH, nBatches - b0);
  if (kmax <= 0) return;

  stage_batch(sbuf, 0, b0, pts, joints, n);

  for (int i = 0; i < kmax; ++i) {
    const int bi = b0 + i;
    const bool haveNext = (i + 1) < kmax;
    if (haveNext) stage_batch(sbuf, (i + 1) & 1, bi + 1, pts, joints, n);
#ifdef USE_TDM
    // batch bi's 2 DMAs done when <=2 remain (batch bi+1 still in flight).
    // SIMM16 must be a literal immediate -> branch to constant calls.
    if (tid < 32) {
      if (haveNext) __builtin_amdgcn_s_wait_tensorcnt(2);
      else          __builtin_amdgcn_s_wait_tensorcnt(0);
    }
#endif
    __syncthreads();

    const float* bp = sbuf[i & 1];
#pragma unroll 2
    for (int j = 0; j < PPW2; ++j) {
      const int pl = wave * PPW2 + j;
      const int p  = bi * BATCH + pl;
      if (p >= n) break;  // wave-uniform

      const float q0 = (bp[3 * pl + 0] - 1.3f) * SCALE - 1.0f;
      const float q1 = (bp[3 * pl + 1] - 1.3f) * SCALE - 1.0f;
      const float q2 = (bp[3 * pl + 2] - 1.3f) * SCALE - 1.0f;
      const float pj = bp[BATCH * 3 + pl];

      const float y3 = fminf(fmaxf((pj + 1.0f) * 0.5f * 99.0f, 0.0f), 99.0f);
      const float w3 = y3 - floorf(y3);

      float* outp = out + (size_t)p * 128 + lane;

#pragma unroll
      for (int si = 0; si < 4; ++si) {
        const int R = 64 << si;
        const float Rm1 = (float)(R - 1);

        const float t0 = fminf(fmaxf((q0 + 1.0f) * 0.5f * Rm1, 0.0f), Rm1);
        const float t1 = fminf(fmaxf((q1 + 1.0f) * 0.5f * Rm1, 0.0f), Rm1);
        const float t2 = fminf(fmaxf((q2 + 1.0f) * 0.5f * Rm1, 0.0f), Rm1);
        const int a0 = (int)t0, b0i = (int)t1, c0 = (int)t2;
        const int a1 = min(a0 + 1, R - 1);
        const int b1 = min(b0i + 1, R - 1);
        const int c1 = min(c0 + 1, R - 1);
        const float u0 = t0 - (float)a0;
        const float u1 = t1 - (float)b0i;
        const float u2 = t2 - (float)c0;

        const float P01 = bilin_eval<TRANSPOSED>(gp.g[si * 3 + 0], lane, R, a0, a1, u0, b0i, b1, u1);
        const float P02 = bilin_eval<TRANSPOSED>(gp.g[si * 3 + 1], lane, R, a0, a1, u0, c0, c1, u2);
        const float P12 = bilin_eval<TRANSPOSED>(gp.g[si * 3 + 2], lane, R, b0i, b1, u1, c0, c1, u2);

        const float A0 = (1.0f - u0) + u0;
        const float A1 = (1.0f - u1) + u1;
        const float A2 = (1.0f - u2) + u2;
        const float f0 = A0 * (1.0f - w3) + A0 * w3;
        const float f1 = A1 * (1.0f - w3) + A1 * w3;
        const float f2 = A2 * (1.0f - w3) + A2 * w3;

        // reference multiply order: (0,1),(0,2),(0,3),(1,2),(1,3),(2,3)
        float prod = P01 * P02;
        prod *= f0;
        prod *= P12;
        prod *= f1;
        prod *= f2;

        __builtin_nontemporal_store(prod, outp + si * 32);
      }
    }
    __syncthreads();  // all readers done before buf (i&1) is re-staged
  }
}

// ------------------------------- launcher ----------------------------------
extern "C" void kernel_launch(void* const* d_in, const int* in_sizes, int n_in,
                              void* d_out, int out_size, void* d_ws, size_t ws_size,
                              hipStream_t stream) {
  const float* pts    = (const float*)d_in[0];
  const float* joints = (const float*)d_in[1];
  float* out = (float*)d_out;
  const int n = in_sizes[0] / 3;

  static const int real_ci[3] = {0, 1, 3};  // combos (0,1),(0,2),(1,2)

  size_t offs[12];
  size_t total_f = 0;
  for (int si = 0; si < 4; ++si) {
    const size_t R = (size_t)64 << si;
    for (int k = 0; k < 3; ++k) {
      offs[si * 3 + k] = total_f;
      total_f += R * R * 32;
    }
  }
  const bool useT = (d_ws != nullptr) && (ws_size >= total_f * sizeof(float));

  GridPtrs gp;
  if (useT) {
    float* wsf = (float*)d_ws;
    for (int si = 0; si < 4; ++si) {
      const int R = 64 << si;
      for (int k = 0; k < 3; ++k) {
        const float* src = (const float*)d_in[2 + si * 6 + real_ci[k]];
        float* dst = wsf + offs[si * 3 + k];
        dim3 tg(R / 64, R);
        transpose_chw_to_hwc<<<tg, 256, 0, stream>>>(src, dst, R, R);
        gp.g[si * 3 + k] = dst;
      }
    }
  } else {
    for (int si = 0; si < 4; ++si)
      for (int k = 0; k < 3; ++k)
        gp.g[si * 3 + k] = (const float*)d_in[2 + si * 6 + real_ci[k]];
  }

  const int nBatches = (n + BATCH - 1) / BATCH;
  const int blocks   = (nBatches + KBATCH - 1) / KBATCH;
  if (useT)
    hexplane_kernel<true><<<blocks, 256, 0, stream>>>(pts, joints, out, n, nBatches, gp);
  else
    hexplane_kernel<false><<<blocks, 256, 0, stream>>>(pts, joints, out, n, nBatches, gp);
}